// Encoder_39367670235545
// MI455X (gfx1250) — compile-verified
//
#include <hip/hip_runtime.h>

typedef __attribute__((ext_vector_type(2))) float v2f;
typedef __attribute__((ext_vector_type(8))) float v8f;

#define NNODES 100000
#define NEDGES 600000
#define DIM    128
#define NGRAPH 512

// ---------------------------------------------------------------- zero fill
__global__ void zero_kernel(float* __restrict__ p, int n) {
  int i = blockIdx.x * blockDim.x + threadIdx.x;
  int stride = gridDim.x * blockDim.x;
  for (; i < n; i += stride) p[i] = 0.0f;
}

// ---------------------------------------------------- edge scatter (mean agg)
// one wave32 per edge: lanes each carry float4 (32*4 = 128 features)
__global__ void __launch_bounds__(256) scatter_kernel(
    const float* __restrict__ h,
    const int* __restrict__ src,
    const int* __restrict__ dst,
    float* __restrict__ agg,
    float* __restrict__ deg) {
  int e = blockIdx.x * (blockDim.x >> 5) + (threadIdx.x >> 5);
  if (e >= NEDGES) return;
  int lane = threadIdx.x & 31;
  int s = src[e];
  int d = dst[e];
  const float4 v = ((const float4*)(h + (size_t)s * DIM))[lane];
  float* ap = agg + (size_t)d * DIM + lane * 4;
  atomicAdd(ap + 0, v.x);
  atomicAdd(ap + 1, v.y);
  atomicAdd(ap + 2, v.z);
  atomicAdd(ap + 3, v.w);
  if (lane == 0) atomicAdd(deg + d, 1.0f);
}

// ------------------------------------------------ fused SAGE layer (WMMA f32)
// hout = relu( (agg / max(deg,1)) @ Wl + bl + hin @ Wr )
// One wave per 16x16 output tile; 8 waves per block cover one 16-row slab
// across all 128 output columns. K-loop in steps of 4 via
// V_WMMA_F32_16X16X4_F32, two WMMAs per step sharing the accumulator.
__global__ void __launch_bounds__(256) sage_gemm_kernel(
    const float* __restrict__ agg, const float* __restrict__ deg,
    const float* __restrict__ hin,
    const float* __restrict__ Wl, const float* __restrict__ bl,
    const float* __restrict__ Wr,
    float* __restrict__ hout) {
  const int wave = threadIdx.x >> 5;   // 0..7 -> column tile
  const int lane = threadIdx.x & 31;
  const int half = lane >> 4;          // 0: K=k,k+1   1: K=k+2,k+3
  const int lidx = lane & 15;
  const int row0 = blockIdx.x * 16;
  const int col0 = wave * 16;
  const int r = row0 + lidx;           // A-fragment row owned by this lane
  const int c = col0 + lidx;           // B-fragment column owned by this lane

  const float invdeg = 1.0f / fmaxf(deg[r], 1.0f);
  const float* __restrict__ aggr = agg + (size_t)r * DIM;
  const float* __restrict__ hr   = hin + (size_t)r * DIM;

  v8f acc = {};
  #pragma unroll 4
  for (int k = 0; k < DIM; k += 4) {
    const int ka = k + 2 * half;
    v2f aM, aH, bL, bR;
    // A (16x4 f32): lane row r, K elements ka, ka+1
    aM[0] = aggr[ka]     * invdeg;
    aM[1] = aggr[ka + 1] * invdeg;
    aH[0] = hr[ka];
    aH[1] = hr[ka + 1];
    // B (4x16 f32): lane col c, K elements ka, ka+1
    bL[0] = Wl[(size_t)ka * DIM + c];
    bL[1] = Wl[(size_t)(ka + 1) * DIM + c];
    bR[0] = Wr[(size_t)ka * DIM + c];
    bR[1] = Wr[(size_t)(ka + 1) * DIM + c];
    acc = __builtin_amdgcn_wmma_f32_16x16x4_f32(
        false, aM, false, bL, (short)0, acc, false, false);
    acc = __builtin_amdgcn_wmma_f32_16x16x4_f32(
        false, aH, false, bR, (short)0, acc, false, false);
  }

  // D layout: VGPR j -> row (j + 8*half), col = lidx within tile
  const float bias = bl[c];
  #pragma unroll
  for (int j = 0; j < 8; ++j) {
    const int row = row0 + j + half * 8;
    hout[(size_t)row * DIM + c] = fmaxf(acc[j] + bias, 0.0f);
  }
}

// -------------------------------------------------------- per-graph sum pool
__global__ void __launch_bounds__(256) pool_kernel(
    const float* __restrict__ h,
    const int* __restrict__ batch,
    float* __restrict__ out) {
  int n = blockIdx.x * (blockDim.x >> 5) + (threadIdx.x >> 5);
  if (n >= NNODES) return;
  int lane = threadIdx.x & 31;
  int g = batch[n];
  const float4 v = ((const float4*)(h + (size_t)n * DIM))[lane];
  float* op = out + (size_t)g * DIM + lane * 4;
  atomicAdd(op + 0, v.x);
  atomicAdd(op + 1, v.y);
  atomicAdd(op + 2, v.z);
  atomicAdd(op + 3, v.w);
}

extern "C" void kernel_launch(void* const* d_in, const int* in_sizes, int n_in,
                              void* d_out, int out_size, void* d_ws, size_t ws_size,
                              hipStream_t stream) {
  const float* x     = (const float*)d_in[0];
  const int*   edge  = (const int*)d_in[1];
  const int*   batch = (const int*)d_in[2];
  const float* Wl[3] = {(const float*)d_in[3], (const float*)d_in[6], (const float*)d_in[9]};
  const float* bl[3] = {(const float*)d_in[4], (const float*)d_in[7], (const float*)d_in[10]};
  const float* Wr[3] = {(const float*)d_in[5], (const float*)d_in[8], (const float*)d_in[11]};
  const int* src = edge;           // edge_index[0]
  const int* dst = edge + NEDGES;  // edge_index[1]

  float* ws  = (float*)d_ws;
  float* agg = ws;                                 // [N, D]
  float* deg = agg + (size_t)NNODES * DIM;         // [N] (contiguous after agg)
  float* hA  = deg + NNODES;                       // [N, D]
  float* hB  = hA + (size_t)NNODES * DIM;          // [N, D]
  float* out = (float*)d_out;                      // [G, D]

  const float* hin = x;
  float* hbuf[2] = {hA, hB};
  for (int l = 0; l < 3; ++l) {
    zero_kernel<<<2048, 256, 0, stream>>>(agg, NNODES * DIM + NNODES);
    scatter_kernel<<<(NEDGES + 7) / 8, 256, 0, stream>>>(hin, src, dst, agg, deg);
    float* hout = hbuf[l & 1];
    sage_gemm_kernel<<<NNODES / 16, 256, 0, stream>>>(
        agg, deg, hin, Wl[l], bl[l], Wr[l], hout);
    hin = hout;
  }

  zero_kernel<<<64, 256, 0, stream>>>(out, NGRAPH * DIM);
  pool_kernel<<<(NNODES + 7) / 8, 256, 0, stream>>>(hin, batch, out);
}